// TimeAggregate_68796786147747
// MI455X (gfx1250) — compile-verified
//
#include <hip/hip_runtime.h>

typedef __attribute__((ext_vector_type(2))) float v2f;
typedef __attribute__((ext_vector_type(8))) float v8f;

// ---------------------------------------------------------------------------
// Phase 0: zero the aggregation buffer (graph-safe, runs every call)
// ---------------------------------------------------------------------------
__global__ void zero_f4_kernel(float4* __restrict__ p, int n4) {
    int i = blockIdx.x * blockDim.x + threadIdx.x;
    if (i < n4) p[i] = make_float4(0.f, 0.f, 0.f, 0.f);
}

// ---------------------------------------------------------------------------
// Phase 1: scatter-add edge features into destination-node buckets.
// One wave (32 lanes) per edge; each lane handles a contiguous float4 of the
// 128-wide feature row -> coalesced 128B*4 loads, hw fp32 atomics into an
// L2-resident 51MB accumulator (global_atomic_add_f32).
// ---------------------------------------------------------------------------
__global__ __launch_bounds__(256)
void scatter_kernel(const float* __restrict__ te, const int* __restrict__ col,
                    float* __restrict__ agg, int E) {
    int gid  = blockIdx.x * blockDim.x + threadIdx.x;
    int edge = gid >> 5;
    int lane = gid & 31;
    if (edge >= E) return;                       // wave-uniform branch

    int c = __builtin_amdgcn_readfirstlane(col[edge]);  // uniform dest node

    const float4* src = (const float4*)(te + (size_t)edge * 128);
    float4 v = src[lane];

    float* dst = agg + (size_t)c * 128 + lane * 4;
    unsafeAtomicAdd(dst + 0, v.x);
    unsafeAtomicAdd(dst + 1, v.y);
    unsafeAtomicAdd(dst + 2, v.z);
    unsafeAtomicAdd(dst + 3, v.w);
}

// ---------------------------------------------------------------------------
// Phase 2: out = agg @ W.T + b  using full-precision V_WMMA_F32_16X16X4_F32.
//  - block = 256 threads = 8 waves; block covers 16 rows x 256 cols
//  - wave w covers N-tiles [32w, 32w+16) and [32w+16, 32w+32), sharing A frag
//  - K = 128 -> 32 WMMA steps per N-tile
// Fragment mapping (ISA 7.12.2, f32):
//   A (16x4):  lane = (k>=2)*16 + m, vgpr = k&1 -> per-lane float2 at
//              agg[row, kbase + 2*(lane>>4)]
//   B (4x16):  B[k][n] = W[n][k]; lane n = lane&15, khalf = lane>>4 ->
//              per-lane float2 at W[nbase + n, kbase + 2*khalf]
//   C/D (16x16): vgpr i, lane l -> m = (l>>4)*8 + i, n = l&15
// ---------------------------------------------------------------------------
__global__ __launch_bounds__(256)
void gemm_wmma_f32_kernel(const float* __restrict__ agg,
                          const float* __restrict__ W,
                          const float* __restrict__ bias,
                          float* __restrict__ out, int Nrows) {
    const int lane  = threadIdx.x & 31;
    const int wave  = threadIdx.x >> 5;     // 0..7
    const int mbase = blockIdx.x << 4;      // 16 rows per block
    const int n0    = wave << 5;            // 32 cols per wave
    const int nlow  = lane & 15;
    const int khalf = lane >> 4;            // 0 or 1

    // A-fragment row for this lane (clamp for the M tail; WMMA needs full EXEC)
    int arow = mbase + nlow;
    if (arow >= Nrows) arow = Nrows - 1;

    const float* aptr  = agg + (size_t)arow * 128 + (khalf << 1);
    const float* wptr0 = W + (size_t)(n0 + nlow) * 128 + (khalf << 1);
    const float* wptr1 = wptr0 + (size_t)16 * 128;

    v8f c0 = {};
    v8f c1 = {};

#pragma unroll 8
    for (int k = 0; k < 128; k += 4) {
        v2f a  = *(const v2f*)(aptr  + k);
        v2f b0 = *(const v2f*)(wptr0 + k);
        v2f b1 = *(const v2f*)(wptr1 + k);
        // (neg_a, A, neg_b, B, c_mod, C, reuse_a, reuse_b)
        c0 = __builtin_amdgcn_wmma_f32_16x16x4_f32(false, a, false, b0,
                                                   (short)0, c0, false, false);
        c1 = __builtin_amdgcn_wmma_f32_16x16x4_f32(false, a, false, b1,
                                                   (short)0, c1, false, false);
    }

    const float bias0 = bias[n0 + nlow];
    const float bias1 = bias[n0 + 16 + nlow];

#pragma unroll
    for (int i = 0; i < 8; ++i) {
        int mrow = mbase + (khalf << 3) + i;
        if (mrow < Nrows) {
            size_t r = (size_t)mrow * 256;
            out[r + n0 + nlow]      = c0[i] + bias0;
            out[r + n0 + 16 + nlow] = c1[i] + bias1;
        }
    }
}

// ---------------------------------------------------------------------------
// Inputs (setup_inputs order): edge_index [2,E] i32, time_encoding [E,128] f32,
// num_nodes (device scalar, unused), W [256,128] f32, b [256] f32.
// Output: [N,256] f32.  Workspace: agg [N,128] f32 (51.2 MB).
// ---------------------------------------------------------------------------
extern "C" void kernel_launch(void* const* d_in, const int* in_sizes, int n_in,
                              void* d_out, int out_size, void* d_ws, size_t ws_size,
                              hipStream_t stream) {
    const int*   edge_index = (const int*)d_in[0];
    const float* te         = (const float*)d_in[1];
    const float* W          = (const float*)d_in[3];
    const float* bias       = (const float*)d_in[4];
    float*       out        = (float*)d_out;

    const int E     = in_sizes[0] / 2;        // 1,600,000
    const int OUT   = in_sizes[4];            // 256
    const int IN    = in_sizes[3] / OUT;      // 128
    const int Nrows = out_size / OUT;         // 100,000
    (void)IN; (void)n_in; (void)ws_size;

    float* agg = (float*)d_ws;                // Nrows*128 floats

    // Phase 0: zero agg
    int n4 = Nrows * 128 / 4;
    zero_f4_kernel<<<(n4 + 255) / 256, 256, 0, stream>>>((float4*)agg, n4);

    // Phase 1: scatter-add (one wave per edge)
    const int* col = edge_index + E;          // edge_index[1] row
    long threads_total = (long)E * 32;
    int blocks = (int)((threads_total + 255) / 256);
    scatter_kernel<<<blocks, 256, 0, stream>>>(te, col, agg, E);

    // Phase 2: fp32 WMMA GEMM + bias
    gemm_wmma_f32_kernel<<<(Nrows + 15) / 16, 256, 0, stream>>>(agg, W, bias, out, Nrows);
}